// Block_51883204936056
// MI455X (gfx1250) — compile-verified
//
#include <hip/hip_runtime.h>

// ---------------------------------------------------------------------------
// CDNA5 types / feature guards
// ---------------------------------------------------------------------------
typedef __attribute__((ext_vector_type(2))) float v2f;
typedef __attribute__((ext_vector_type(8))) float v8f;

#if defined(__gfx1250__) && __has_builtin(__builtin_amdgcn_global_load_async_to_lds_b32) && __has_builtin(__builtin_amdgcn_s_wait_asynccnt)
#define HAS_ASYNC_LDS 1
#else
#define HAS_ASYNC_LDS 0
#endif

typedef __attribute__((address_space(1))) int as1_int;
typedef __attribute__((address_space(3))) int as3_int;

__device__ __forceinline__ void stage_word(const float* g, float* l, bool inb) {
  if (inb) {
#if HAS_ASYNC_LDS
    __builtin_amdgcn_global_load_async_to_lds_b32(
        (as1_int*)g, (as3_int*)l, 0, 0);
#else
    *l = *g;
#endif
  } else {
    *l = 0.0f;
  }
}

__device__ __forceinline__ void stage_fence() {
#if HAS_ASYNC_LDS
  __builtin_amdgcn_s_wait_asynccnt(0);
#endif
}

// ---------------------------------------------------------------------------
// Stage 1: fused 5x (conv3x3 + bias + BU + clamp) per pyramid level.
// One block = one 64x64 output tile. Computes an 80x80 region each iteration
// as a 5x5 grid of 16x16 WMMA tiles; central 64x64 is exact after 5 iters.
// Region coords: r in [-1,80] -> LDS row rr=r+1 (82 rows); c in [-1,82] ->
// LDS col cc=c+1 (84 cols, stride 84 is bank-conflict-free for our patterns).
// ---------------------------------------------------------------------------
#define RSTRIDE 84
#define RROWS   82
#define CREG    80   // computed region per iteration

__global__ __launch_bounds__(256) void pyr_iter_kernel(
    const float* __restrict__ in, const float* __restrict__ BU,
    const float* __restrict__ wconv, const float* __restrict__ bias,
    float* __restrict__ out, int H, int W)
{
  __shared__ float bufA[RROWS * RSTRIDE];
  __shared__ float bufB[RROWS * RSTRIDE];
  __shared__ float bus [CREG  * RSTRIDE];

  const int tid  = threadIdx.x;
  const int lane = tid & 31;
  const int wv   = tid >> 5;      // wave id 0..7
  const int n    = lane & 15;     // N (and A's M) index within wave
  const int kh   = lane >> 4;     // K-half selector

  const int tx = blockIdx.x, ty = blockIdx.y, bimg = blockIdx.z;
  const int gx0 = tx * 64 - 8;    // image col of region col 0
  const int gy0 = ty * 64 - 8;    // image row of region row 0
  const long long planeOff = (long long)bimg * H * W;
  const float* inP = in + planeOff;
  const float* buP = BU + planeOff;

  // conv weights (uniform)
  float K[9];
#pragma unroll
  for (int i = 0; i < 9; ++i) K[i] = wconv[i];
  const float bsv = bias[0];

  // ---- stage input (full 82x84, zero-fill outside image) + zero bufB ----
  for (int idx = tid; idx < RROWS * RSTRIDE; idx += 256) {
    int rr = idx / RSTRIDE, cc = idx % RSTRIDE;
    int iy = gy0 + rr - 1, ix = gx0 + cc - 1;
    bool inb = (iy >= 0) & (iy < H) & (ix >= 0) & (ix < W);
    stage_word(inP + (long long)iy * W + ix, &bufA[idx], inb);
    bufB[idx] = 0.0f;
  }
  // ---- stage BU over region rows/cols 0..79 ----
  for (int idx = tid; idx < CREG * CREG; idx += 256) {
    int rr = idx / CREG, cc = idx % CREG;
    int iy = gy0 + rr, ix = gx0 + cc;
    bool inb = (iy >= 0) & (iy < H) & (ix >= 0) & (ix < W);
    stage_word(buP + (long long)iy * W + ix, &bus[rr * RSTRIDE + cc], inb);
  }
  stage_fence();
  __syncthreads();

  // ---- precompute banded B matrices: B(k,n) = K[dr][p-n], p = c0+k ----
  // B layout (symmetric to A): lane n = N, vgpr v -> K row = 2*kh + v.
  v2f Bm[15];
#pragma unroll
  for (int dr = 0; dr < 3; ++dr) {
#pragma unroll
    for (int ch = 0; ch < 5; ++ch) {
      v2f b;
#pragma unroll
      for (int v = 0; v < 2; ++v) {
        int p = ch * 4 + 2 * kh + v;
        int j = p - n;
        float val = (j >= 0 && j <= 2) ? K[dr * 3 + j] : 0.0f;
        if (v == 0) b.x = val; else b.y = val;
      }
      Bm[dr * 5 + ch] = b;
    }
  }

  float* cur = bufA;
  float* nxt = bufB;
#pragma unroll 1
  for (int it = 0; it < 5; ++it) {
    // 25 WMMA tiles, strided over the 8 waves (wave-uniform; EXEC all ones)
    for (int t = wv; t < 25; t += 8) {
      const int row0 = (t / 5) * 16;   // region row of tile origin
      const int col0 = (t % 5) * 16;   // region col of tile origin

      // C = bias + BU  (C/D layout: M = v + 8*kh, N = n)
      v8f acc;
#pragma unroll
      for (int v = 0; v < 8; ++v)
        acc[v] = bsv + bus[(row0 + v + 8 * kh) * RSTRIDE + col0 + n];

      // conv3x3 = sum over 3 kernel rows of banded matmuls, K split in 5 chunks
#pragma unroll
      for (int dr = 0; dr < 3; ++dr) {
#pragma unroll
        for (int ch = 0; ch < 5; ++ch) {
          // A layout: M = n (lane%16), K = 2*kh + v ; input col = col0+c0-1+K
          const int rr = row0 + n + dr;             // buf row (region row +1)
          const int cc = col0 + ch * 4 + 2 * kh;    // buf col of K element 0
          v2f a;
          a.x = cur[rr * RSTRIDE + cc];
          a.y = cur[rr * RSTRIDE + cc + 1];
          acc = __builtin_amdgcn_wmma_f32_16x16x4_f32(
              false, a, false, Bm[dr * 5 + ch], (short)0, acc, false, false);
        }
      }

      // clamp, re-impose zero padding outside image, store to other buffer
#pragma unroll
      for (int v = 0; v < 8; ++v) {
        int ry = row0 + v + 8 * kh;   // region row
        int rx = col0 + n;            // region col
        int iy = gy0 + ry, ix = gx0 + rx;
        float val = fminf(fmaxf(acc[v], -1.0f), 1.0f);
        val = ((iy >= 0) & (iy < H) & (ix >= 0) & (ix < W)) ? val : 0.0f;
        nxt[(ry + 1) * RSTRIDE + (rx + 1)] = val;
      }
    }
    __syncthreads();
    float* tmpp = cur; cur = nxt; nxt = tmpp;
  }

  // ---- write exact central 64x64 (region rows/cols 8..71 -> buf 9..72) ----
  float* outP = out + planeOff;
  for (int idx = tid; idx < 64 * 64; idx += 256) {
    int oy = idx >> 6, ox = idx & 63;
    outP[(long long)(ty * 64 + oy) * W + (tx * 64 + ox)] =
        cur[(9 + oy) * RSTRIDE + (9 + ox)];
  }
}

// ---------------------------------------------------------------------------
// Stage 2: cross-scale fusion (1x1 conv over 3 resampled channels)
// x5:[B,H,W]  d5:[B,H/2,W/2]  u5:[B,2H,2W]
// ---------------------------------------------------------------------------
__global__ __launch_bounds__(256) void fuse_x_kernel(
    const float* __restrict__ x5, const float* __restrict__ d5,
    const float* __restrict__ u5, const float* __restrict__ wmix,
    float* __restrict__ out, int B, int H, int W)
{
  const float w0 = wmix[0], w1 = wmix[1], w2 = wmix[2];
  const long long plane = (long long)H * W;
  const long long total = (long long)B * plane;
  for (long long i = (long long)blockIdx.x * blockDim.x + threadIdx.x;
       i < total; i += (long long)gridDim.x * blockDim.x) {
    int b = (int)(i / plane);
    int rem = (int)(i - (long long)b * plane);
    int h = rem / W, w = rem % W;
    const float* up = u5 + (long long)b * 4 * plane;
    const int W2 = 2 * W;
    float u2 = 0.25f * (up[(2 * h) * W2 + 2 * w]     + up[(2 * h) * W2 + 2 * w + 1] +
                        up[(2 * h + 1) * W2 + 2 * w] + up[(2 * h + 1) * W2 + 2 * w + 1]);
    float dv = d5[(long long)b * (plane / 4) + (h >> 1) * (W / 2) + (w >> 1)];
    out[i] = w0 * u2 + w1 * x5[i] + w2 * dv;
  }
}

__global__ __launch_bounds__(256) void fuse_down_kernel(
    const float* __restrict__ x5, const float* __restrict__ d5,
    const float* __restrict__ u5, const float* __restrict__ wmix,
    float* __restrict__ out, int B, int H, int W)
{
  const float w0 = wmix[0], w1 = wmix[1], w2 = wmix[2];
  const int Hd = H / 2, Wd = W / 2;
  const long long plane = (long long)Hd * Wd;
  const long long total = (long long)B * plane;
  for (long long i = (long long)blockIdx.x * blockDim.x + threadIdx.x;
       i < total; i += (long long)gridDim.x * blockDim.x) {
    int b = (int)(i / plane);
    int rem = (int)(i - (long long)b * plane);
    int r = rem / Wd, s = rem % Wd;
    const float* up = u5 + (long long)b * 4 * H * W;
    const float* xp = x5 + (long long)b * H * W;
    const int W2 = 2 * W;
    float u4 = 0.0f;
#pragma unroll
    for (int ii = 0; ii < 4; ++ii)
#pragma unroll
      for (int jj = 0; jj < 4; ++jj)
        u4 += up[(4 * r + ii) * W2 + 4 * s + jj];
    u4 *= (1.0f / 16.0f);
    float x2 = 0.25f * (xp[(2 * r) * W + 2 * s]     + xp[(2 * r) * W + 2 * s + 1] +
                        xp[(2 * r + 1) * W + 2 * s] + xp[(2 * r + 1) * W + 2 * s + 1]);
    out[i] = w0 * u4 + w1 * x2 + w2 * d5[i];
  }
}

__global__ __launch_bounds__(256) void fuse_up_kernel(
    const float* __restrict__ x5, const float* __restrict__ d5,
    const float* __restrict__ u5, const float* __restrict__ wmix,
    float* __restrict__ out, int B, int H, int W)
{
  const float w0 = wmix[0], w1 = wmix[1], w2 = wmix[2];
  const int H2 = 2 * H, W2 = 2 * W;
  const long long plane = (long long)H2 * W2;
  const long long total = (long long)B * plane;
  for (long long i = (long long)blockIdx.x * blockDim.x + threadIdx.x;
       i < total; i += (long long)gridDim.x * blockDim.x) {
    int b = (int)(i / plane);
    int rem = (int)(i - (long long)b * plane);
    int p = rem / W2, q = rem % W2;
    float xv = x5[(long long)b * H * W + (p >> 1) * W + (q >> 1)];
    float dv = d5[(long long)b * (H / 2) * (W / 2) + (p >> 2) * (W / 2) + (q >> 2)];
    out[i] = w0 * u5[i] + w1 * xv + w2 * dv;
  }
}

// ---------------------------------------------------------------------------
// Launcher
// ---------------------------------------------------------------------------
extern "C" void kernel_launch(void* const* d_in, const int* in_sizes, int n_in,
                              void* d_out, int out_size, void* d_ws, size_t ws_size,
                              hipStream_t stream) {
  const float* x   = (const float*)d_in[0];
  const float* dn  = (const float*)d_in[1];
  const float* up  = (const float*)d_in[2];
  const float* BUx = (const float*)d_in[3];
  const float* BUd = (const float*)d_in[4];
  const float* BUu = (const float*)d_in[5];
  const float* wAx = (const float*)d_in[6];
  const float* bx  = (const float*)d_in[7];
  const float* wAd = (const float*)d_in[8];
  const float* bd  = (const float*)d_in[9];
  const float* wAu = (const float*)d_in[10];
  const float* bu  = (const float*)d_in[11];
  const float* w1x = (const float*)d_in[12];
  const float* w1d = (const float*)d_in[13];
  const float* w1u = (const float*)d_in[14];

  const int H = 512, W = 512;
  const int B = in_sizes[0] / (H * W);

  // workspace: stage-1 results (x5, d5, u5), fp32
  float* x5 = (float*)d_ws;
  float* d5 = x5 + (size_t)B * H * W;
  float* u5 = d5 + (size_t)B * (H / 2) * (W / 2);

  float* xo = (float*)d_out;
  float* do_ = xo + (size_t)B * H * W;
  float* uo = do_ + (size_t)B * (H / 2) * (W / 2);

  dim3 blk(256);
  pyr_iter_kernel<<<dim3(W / 64, H / 64, B), blk, 0, stream>>>(x,  BUx, wAx, bx, x5, H, W);
  pyr_iter_kernel<<<dim3(W / 128, H / 128, B), blk, 0, stream>>>(dn, BUd, wAd, bd, d5, H / 2, W / 2);
  pyr_iter_kernel<<<dim3(W / 32, H / 32, B), blk, 0, stream>>>(up, BUu, wAu, bu, u5, 2 * H, 2 * W);

  fuse_x_kernel   <<<dim3(2048), blk, 0, stream>>>(x5, d5, u5, w1x, xo,  B, H, W);
  fuse_down_kernel<<<dim3(512),  blk, 0, stream>>>(x5, d5, u5, w1d, do_, B, H, W);
  fuse_up_kernel  <<<dim3(4096), blk, 0, stream>>>(x5, d5, u5, w1u, uo,  B, H, W);
}